// Model_10299331575985
// MI455X (gfx1250) — compile-verified
//
#include <hip/hip_runtime.h>

// Three independent streaming reductions; HBM-bound (~640 MiB @ 23.3 TB/s ~ 29 us floor).
// Strategy: b128 non-temporal loads, branchless register top-k, wave32 xor-butterfly merges.

typedef float v4f __attribute__((ext_vector_type(4)));

#define NEG_INF (-__builtin_huge_valf())
#define POS_INF (__builtin_huge_valf())

// ---------- branchless sorted-descending insertion helpers ----------

// Bubble-insert v into (a0 >= a1 >= a2 >= a3): pure v_max/v_min ladder, no branches.
__device__ __forceinline__ void ins4(float v, float& a0, float& a1, float& a2, float& a3) {
    float t;
    t = fmaxf(a0, v); v = fminf(a0, v); a0 = t;
    t = fmaxf(a1, v); v = fminf(a1, v); a1 = t;
    t = fmaxf(a2, v); v = fminf(a2, v); a2 = t;
    a3 = fmaxf(a3, v);
}

// Branchless insert of (v,idx) into sorted (v0,i0) >= (v1,i1) >= (v2,i2).
// Compiles to v_cmp + v_cndmask chains; index travels with its value.
__device__ __forceinline__ void ins3(float v, int idx,
                                     float& v0, int& i0,
                                     float& v1, int& i1,
                                     float& v2, int& i2) {
    const bool g0 = v > v0;
    const bool g1 = v > v1;
    const bool g2 = v > v2;
    float nv2 = g1 ? v1 : (g2 ? v : v2);
    int   ni2 = g1 ? i1 : (g2 ? idx : i2);
    float nv1 = g0 ? v0 : (g1 ? v : v1);
    int   ni1 = g0 ? i0 : (g1 ? idx : i1);
    float nv0 = g0 ? v : v0;
    int   ni0 = g0 ? idx : i0;
    v0 = nv0; i0 = ni0;
    v1 = nv1; i1 = ni1;
    v2 = nv2; i2 = ni2;
}

// ---------- kernel 1: top-4 of 1024-float rows, one wave32 per row ----------

__global__ void topk4_x(const float* __restrict__ x, float* __restrict__ out, int nrows) {
    const int wid  = (blockIdx.x * blockDim.x + threadIdx.x) >> 5;
    const int lane = threadIdx.x & 31;
    if (wid >= nrows) return;

    const v4f* row = (const v4f*)(x + (size_t)wid * 1024);  // 256 x float4
    float a0 = NEG_INF, a1 = NEG_INF, a2 = NEG_INF, a3 = NEG_INF;

#pragma unroll
    for (int i = 0; i < 8; ++i) {
        // lanes 0..31 read contiguous 512B per iteration -> global_load_b128 NT
        v4f v = __builtin_nontemporal_load(row + lane + i * 32);
        ins4(v[0], a0, a1, a2, a3);
        ins4(v[1], a0, a1, a2, a3);
        ins4(v[2], a0, a1, a2, a3);
        ins4(v[3], a0, a1, a2, a3);
    }

#pragma unroll
    for (int off = 16; off >= 1; off >>= 1) {
        float b0 = __shfl_xor(a0, off, 32);
        float b1 = __shfl_xor(a1, off, 32);
        float b2 = __shfl_xor(a2, off, 32);
        float b3 = __shfl_xor(a3, off, 32);
        ins4(b0, a0, a1, a2, a3);
        ins4(b1, a0, a1, a2, a3);
        ins4(b2, a0, a1, a2, a3);
        ins4(b3, a0, a1, a2, a3);
    }

    if (lane == 0) {
        float* o = out + (size_t)wid * 4;
        o[0] = a0; o[1] = a1; o[2] = a2; o[3] = a3;
    }
}

// ---------- kernel 2: min over dim2 (2048) of y[8,32,2048,64] ----------
// block = 256 threads = 16 float4-columns x 16 k-groups; grid = (256 ab, 8 ksplit).
// Every thread loads v4f -> global_load_b128 NT; each block iteration reads 4 KiB
// contiguous. Per-block LDS tree over k-groups, partials to d_ws (no atomics).

__global__ void ymin_partial(const float* __restrict__ y, float* __restrict__ ws) {
    const int ab  = blockIdx.x;        // 0..255  (a*32+b)
    const int ks  = blockIdx.y;        // 0..7    (k split)
    const int tid = threadIdx.x;       // 0..255
    const int c4  = tid & 15;          // float4 column within 64-float row
    const int g   = tid >> 4;          // 0..15 k-group

    const v4f* base = (const v4f*)(y + (size_t)ab * 2048 * 64) + c4;  // k stride = 16 v4f
    v4f m = {POS_INF, POS_INF, POS_INF, POS_INF};
    const int k0 = ks * 256 + g;
#pragma unroll
    for (int j = 0; j < 16; ++j) {
        v4f v = __builtin_nontemporal_load(base + (size_t)(k0 + j * 16) * 16);
        m[0] = fminf(m[0], v[0]);
        m[1] = fminf(m[1], v[1]);
        m[2] = fminf(m[2], v[2]);
        m[3] = fminf(m[3], v[3]);
    }

    __shared__ v4f s[256];
    s[tid] = m;
    __syncthreads();
#pragma unroll
    for (int st = 8; st >= 1; st >>= 1) {
        if (g < st) {
            v4f o = s[tid + st * 16];
            v4f mm = s[tid];
            mm[0] = fminf(mm[0], o[0]);
            mm[1] = fminf(mm[1], o[1]);
            mm[2] = fminf(mm[2], o[2]);
            mm[3] = fminf(mm[3], o[3]);
            s[tid] = mm;
        }
        __syncthreads();
    }
    if (g == 0) {
        ((v4f*)ws)[((size_t)ab * 8 + ks) * 16 + c4] = s[c4];
    }
}

__global__ void ymin_final(const float* __restrict__ ws, float* __restrict__ out) {
    const int t = blockIdx.x * blockDim.x + threadIdx.x;  // 16384 outputs
    if (t >= 16384) return;
    const int ab = t >> 6, c = t & 63;
    float m = POS_INF;
#pragma unroll
    for (int ks = 0; ks < 8; ++ks)
        m = fminf(m, ws[((size_t)ab * 8 + ks) * 64 + c]);
    out[t] = m;
}

// ---------- kernel 3: top-3 (value,index) of 512-float rows, one wave32 per row ----

__global__ void topk3_z(const float* __restrict__ z,
                        float* __restrict__ outv, float* __restrict__ outi, int nrows) {
    const int wid  = (blockIdx.x * blockDim.x + threadIdx.x) >> 5;
    const int lane = threadIdx.x & 31;
    if (wid >= nrows) return;

    const v4f* row = (const v4f*)(z + (size_t)wid * 512);  // 128 x float4
    float v0 = NEG_INF, v1 = NEG_INF, v2 = NEG_INF;
    int   i0 = 0, i1 = 0, i2 = 0;

#pragma unroll
    for (int i = 0; i < 4; ++i) {
        const int fi = lane + i * 32;           // float4 index
        v4f v = __builtin_nontemporal_load(row + fi);
        const int e = fi * 4;
        ins3(v[0], e + 0, v0, i0, v1, i1, v2, i2);
        ins3(v[1], e + 1, v0, i0, v1, i1, v2, i2);
        ins3(v[2], e + 2, v0, i0, v1, i1, v2, i2);
        ins3(v[3], e + 3, v0, i0, v1, i1, v2, i2);
    }

#pragma unroll
    for (int off = 16; off >= 1; off >>= 1) {
        float b0 = __shfl_xor(v0, off, 32); int j0 = __shfl_xor(i0, off, 32);
        float b1 = __shfl_xor(v1, off, 32); int j1 = __shfl_xor(i1, off, 32);
        float b2 = __shfl_xor(v2, off, 32); int j2 = __shfl_xor(i2, off, 32);
        ins3(b0, j0, v0, i0, v1, i1, v2, i2);
        ins3(b1, j1, v0, i0, v1, i1, v2, i2);
        ins3(b2, j2, v0, i0, v1, i1, v2, i2);
    }

    if (lane == 0) {
        float* ov = outv + (size_t)wid * 3;
        float* oi = outi + (size_t)wid * 3;
        ov[0] = v0; ov[1] = v1; ov[2] = v2;
        oi[0] = (float)i0; oi[1] = (float)i1; oi[2] = (float)i2;
    }
}

// ---------- launch ----------

extern "C" void kernel_launch(void* const* d_in, const int* in_sizes, int n_in,
                              void* d_out, int out_size, void* d_ws, size_t ws_size,
                              hipStream_t stream) {
    const float* x = (const float*)d_in[0];  // [32,2048,1024]
    const float* y = (const float*)d_in[1];  // [8,32,2048,64]
    const float* z = (const float*)d_in[2];  // [32,4096,512]

    float* out = (float*)d_out;
    float* out_x  = out;                       // 32*2048*4  = 262144
    float* out_y  = out_x + 262144;            // 8*32*64    = 16384
    float* out_zv = out_y + 16384;             // 32*4096*3  = 393216
    float* out_zi = out_zv + 393216;           // 32*4096*3  = 393216

    float* ws = (float*)d_ws;                  // 256*8*64 floats = 512 KiB partials

    // x: 65536 rows, 8 waves per 256-thread block -> 8192 blocks
    topk4_x<<<8192, 256, 0, stream>>>(x, out_x, 65536);

    // y: 256 ab-slices x 8 k-splits, then 8-way final reduce
    ymin_partial<<<dim3(256, 8), 256, 0, stream>>>(y, ws);
    ymin_final<<<64, 256, 0, stream>>>(ws, out_y);

    // z: 131072 rows -> 16384 blocks
    topk3_z<<<16384, 256, 0, stream>>>(z, out_zv, out_zi, 131072);
}